// CoresLoss_1357209665715
// MI455X (gfx1250) — compile-verified
//
#include <hip/hip_runtime.h>

#define LOG2E 1.4426950408889634f
#define LN2   0.6931471805599453f

__device__ __forceinline__ float wave_sum(float v) {
#pragma unroll
    for (int o = 16; o > 0; o >>= 1) v += __shfl_xor(v, o, 32);
    return v;
}
__device__ __forceinline__ float wave_max(float v) {
#pragma unroll
    for (int o = 16; o > 0; o >>= 1) v = fmaxf(v, __shfl_xor(v, o, 32));
    return v;
}

// One block per row. 128 threads (4 wave32s); threads 0..124 each own two
// float4s (8 floats) of the 1000-wide row in registers. Single HBM pass;
// exp() results are kept in registers and reused for the log(p+eps) pass.
__global__ __launch_bounds__(128)
void cores_rows(const float* __restrict__ pred,
                const int*   __restrict__ labels,
                const int*   __restrict__ epoch_p,
                float* __restrict__ row_masked,
                float* __restrict__ row_v,
                int C)
{
    const int r    = blockIdx.x;
    const int tid  = threadIdx.x;
    const int lane = tid & 31;
    const int wave = tid >> 5;
    const float* row = pred + (size_t)r * (size_t)C;

    __shared__ float s_red[4];
    __shared__ float s_val;
    __shared__ float s_xlab;

    // gfx1250 prefetch: WGP scope (locality 3) pulls the row into all cache
    // levels. 32 lanes x 128B = 4096B covers the 4000B row.
    if (tid < 32) __builtin_prefetch(row + tid * 32, 0, 3);

    const int nvec = C >> 3;  // 125 threads each handle 2 float4s
    const float NEG_INF = -__builtin_inff();
    float4 x0 = make_float4(NEG_INF, NEG_INF, NEG_INF, NEG_INF);
    float4 x1 = x0;
    if (tid < nvec) {
        const float4* p = reinterpret_cast<const float4*>(row);
        x0 = p[2 * tid + 0];
        x1 = p[2 * tid + 1];
    }

    // ---- pass 1: row max ----
    float m = fmaxf(fmaxf(fmaxf(x0.x, x0.y), fmaxf(x0.z, x0.w)),
                    fmaxf(fmaxf(x1.x, x1.y), fmaxf(x1.z, x1.w)));
    m = wave_max(m);
    if (lane == 0) s_red[wave] = m;
    __syncthreads();
    if (wave == 0) {
        float w = (lane < 4) ? s_red[lane] : NEG_INF;
        w = wave_max(w);
        if (lane == 0) s_val = w;
    }
    __syncthreads();
    m = s_val;
    __syncthreads();

    // ---- pass 2: sum of exp(x - m); keep e in registers (2 indep chains) ----
    float4 e0, e1;
    e0.x = __builtin_amdgcn_exp2f((x0.x - m) * LOG2E);
    e1.x = __builtin_amdgcn_exp2f((x1.x - m) * LOG2E);
    e0.y = __builtin_amdgcn_exp2f((x0.y - m) * LOG2E);
    e1.y = __builtin_amdgcn_exp2f((x1.y - m) * LOG2E);
    e0.z = __builtin_amdgcn_exp2f((x0.z - m) * LOG2E);
    e1.z = __builtin_amdgcn_exp2f((x1.z - m) * LOG2E);
    e0.w = __builtin_amdgcn_exp2f((x0.w - m) * LOG2E);
    e1.w = __builtin_amdgcn_exp2f((x1.w - m) * LOG2E);
    float s = (tid < nvec)
        ? ((e0.x + e0.y + e0.z + e0.w) + (e1.x + e1.y + e1.z + e1.w))
        : 0.0f;
    s = wave_sum(s);
    if (lane == 0) s_red[wave] = s;
    __syncthreads();
    if (wave == 0) {
        float w = (lane < 4) ? s_red[lane] : 0.0f;
        w = wave_sum(w);
        if (lane == 0) s_val = w;
    }
    __syncthreads();
    const float S = s_val;
    __syncthreads();

    // stash x[label] for the ce term
    const int lab = labels[r];
    if (tid == (lab >> 3)) {
        const int c = lab & 7;
        float xv;
        switch (c) {
            case 0: xv = x0.x; break;
            case 1: xv = x0.y; break;
            case 2: xv = x0.z; break;
            case 3: xv = x0.w; break;
            case 4: xv = x1.x; break;
            case 5: xv = x1.y; break;
            case 6: xv = x1.z; break;
            default: xv = x1.w; break;
        }
        s_xlab = xv;
    }

    // ---- pass 3: sum_j log2(p_j + 1e-8), p_j = e_j / S ----
    const float rcpS = 1.0f / S;
    float t = 0.0f;
    if (tid < nvec) {
        float ta = __builtin_amdgcn_logf(e0.x * rcpS + 1e-8f);
        float tb = __builtin_amdgcn_logf(e1.x * rcpS + 1e-8f);
        ta += __builtin_amdgcn_logf(e0.y * rcpS + 1e-8f);
        tb += __builtin_amdgcn_logf(e1.y * rcpS + 1e-8f);
        ta += __builtin_amdgcn_logf(e0.z * rcpS + 1e-8f);
        tb += __builtin_amdgcn_logf(e1.z * rcpS + 1e-8f);
        ta += __builtin_amdgcn_logf(e0.w * rcpS + 1e-8f);
        tb += __builtin_amdgcn_logf(e1.w * rcpS + 1e-8f);
        t = ta + tb;
    }
    t = wave_sum(t);
    __syncthreads();           // protect s_red reuse + ensure s_xlab visible
    if (lane == 0) s_red[wave] = t;
    __syncthreads();
    if (wave == 0) {
        float w = (lane < 4) ? s_red[lane] : 0.0f;
        w = wave_sum(w);
        if (lane == 0) s_val = w;
    }
    __syncthreads();

    if (tid == 0) {
        const float sumlog2 = s_val;                 // sum of log2(p+eps)
        const float mean_neglog = -(sumlog2 * LN2) / (float)C;
        const float lse = m + __builtin_amdgcn_logf(S) * LN2;
        const float ce  = lse - s_xlab;

        const int ep = *epoch_p;
        const float beta = (ep < 20) ? 0.0f
                         : (ep < 80) ? (2.0f * (float)(ep - 20) / 59.0f)
                                     : 2.0f;
        const float loss     = ce - beta * mean_neglog;
        const float loss_sel = ce - mean_neglog;
        const float v = (ep <= 60) ? 1.0f : ((loss_sel <= 0.0f) ? 1.0f : 0.0f);

        row_masked[r] = v * loss;
        row_v[r]      = v;
    }
}

// Single-block deterministic final reduction over B rows.
__global__ __launch_bounds__(1024)
void cores_final(const float* __restrict__ row_masked,
                 const float* __restrict__ row_v,
                 float* __restrict__ out, int B)
{
    const int tid  = threadIdx.x;
    const int lane = tid & 31;
    const int wave = tid >> 5;

    float sm = 0.0f, sv = 0.0f;
    for (int i = tid; i < B; i += 1024) {
        sm += row_masked[i];
        sv += row_v[i];
    }
    sm = wave_sum(sm);
    sv = wave_sum(sv);

    __shared__ float a[32], b[32];
    if (lane == 0) { a[wave] = sm; b[wave] = sv; }
    __syncthreads();
    if (wave == 0) {
        float xs = a[lane];   // exactly 32 waves
        float ys = b[lane];
        xs = wave_sum(xs);
        ys = wave_sum(ys);
        if (lane == 0) {
            // denom==0 -> mean(masked)/1e8, else sum(masked)/denom
            out[0] = (ys == 0.0f) ? (xs / (float)B) * 1e-8f : (xs / ys);
        }
    }
}

extern "C" void kernel_launch(void* const* d_in, const int* in_sizes, int n_in,
                              void* d_out, int out_size, void* d_ws, size_t ws_size,
                              hipStream_t stream)
{
    const float* pred   = (const float*)d_in[0];
    const int*   labels = (const int*)d_in[1];
    const int*   epoch  = (const int*)d_in[2];

    const int B = in_sizes[1];               // 32768
    const int C = in_sizes[0] / B;           // 1000

    float* row_masked = (float*)d_ws;        // [B]
    float* row_v      = row_masked + B;      // [B]

    cores_rows<<<B, 128, 0, stream>>>(pred, labels, epoch, row_masked, row_v, C);
    cores_final<<<1, 1024, 0, stream>>>(row_masked, row_v, (float*)d_out, B);
}